// RankerXQuad_73598559584386
// MI455X (gfx1250) — compile-verified
//
#include <hip/hip_runtime.h>
#include <cstdint>

#define NITEMS 16384
#define TPB    256
#define PER    (NITEMS / TPB)   // 64 items per thread
#define KSEL   10
#define RMAXC  100
#define NBINS  4096
#define CAP    1024
#define LMB    0.4f

// ---- dynamic LDS layout (bytes) ----
#define OFF_SN    0                 // 16384 f32  (64 KB)  score row / normalized scores
#define OFF_HIST  65536             // 4096 u32   (16 KB)  histogram; reused for sorted cands
#define OFF_SVAL  65536             // 100 f32  (aliases hist, used after threshold pass)
#define OFF_SIDX  (65536 + 512)     // 100 i32
#define OFF_OMASK 81920             // 512 u32   (2 KB)    observed bitmask
#define OFF_CVAL  83968             // 1024 f32  (4 KB)    raw candidate values
#define OFF_CIDX  88064             // 1024 i32  (4 KB)    raw candidate indices
#define OFF_REDF  92160             // 256 f32   (1 KB)    reduction scratch
#define OFF_REDI  93184             // 256 i32   (1 KB)    reduction scratch
#define OFF_MISC  94208             // small scalars
#define SMEM_BYTES 94336

struct MiscShm {
  float pickch;
  int   thrbin;
  unsigned ccount;
  int   sel[KSEL];
};

__device__ __forceinline__ float block_reduce_add(float v, float* buf, int tid) {
  __syncthreads();
  buf[tid] = v; __syncthreads();
  for (int s = TPB / 2; s > 0; s >>= 1) {
    if (tid < s) buf[tid] = buf[tid] + buf[tid + s];
    __syncthreads();
  }
  float r = buf[0]; __syncthreads();
  return r;
}
__device__ __forceinline__ float block_reduce_min(float v, float* buf, int tid) {
  __syncthreads();
  buf[tid] = v; __syncthreads();
  for (int s = TPB / 2; s > 0; s >>= 1) {
    if (tid < s) buf[tid] = fminf(buf[tid], buf[tid + s]);
    __syncthreads();
  }
  float r = buf[0]; __syncthreads();
  return r;
}
__device__ __forceinline__ float block_reduce_max(float v, float* buf, int tid) {
  __syncthreads();
  buf[tid] = v; __syncthreads();
  for (int s = TPB / 2; s > 0; s >>= 1) {
    if (tid < s) buf[tid] = fmaxf(buf[tid], buf[tid + s]);
    __syncthreads();
  }
  float r = buf[0]; __syncthreads();
  return r;
}

__global__ __launch_bounds__(TPB)
void rerank_kernel(const float* __restrict__ pred,
                   const float* __restrict__ obs,
                   const int*  __restrict__ head,
                   float* __restrict__ out)
{
  extern __shared__ char smem[];
  float*    sn    = (float*)(smem + OFF_SN);
  unsigned* hist  = (unsigned*)(smem + OFF_HIST);
  float*    sval  = (float*)(smem + OFF_SVAL);
  int*      sidx  = (int*)(smem + OFF_SIDX);
  unsigned* omask = (unsigned*)(smem + OFF_OMASK);
  float*    cval  = (float*)(smem + OFF_CVAL);
  int*      cidx  = (int*)(smem + OFF_CIDX);
  float*    redf  = (float*)(smem + OFF_REDF);
  int*      redi  = (int*)(smem + OFF_REDI);
  MiscShm*  S     = (MiscShm*)(smem + OFF_MISC);

  const int tid = threadIdx.x;
  const int row = blockIdx.x;
  const float* prow = pred + (size_t)row * NITEMS;
  const float* orow = obs  + (size_t)row * NITEMS;
  float*       wrow = out  + (size_t)row * NITEMS;

  // ---- Pass 0: async DMA the score row into LDS (ASYNCcnt path, CDNA5) ----
  {
    unsigned lds_base = (unsigned)(uintptr_t)sn;
    #pragma unroll
    for (int it = 0; it < PER / 4; ++it) {          // 16 x (256 lanes x 16B) = 64 KB
      unsigned byte_off = (unsigned)((it * TPB + tid) * 16);
      unsigned lds_addr = lds_base + byte_off;
      asm volatile("global_load_async_to_lds_b128 %0, %1, %2"
                   :: "v"(lds_addr), "v"(byte_off), "s"(prow)
                   : "memory");
    }
  }

  // zero histogram while the DMA is in flight
  for (int b = tid; b < NBINS; b += TPB) hist[b] = 0u;
  if (tid == 0) S->ccount = 0u;

  // ---- Pass 1 (overlapped): observed stats + bitmask ----
  float osum = 0.f, ohead = 0.f;
  for (int it = 0; it < PER; ++it) {
    int   i = it * TPB + tid;                       // contiguous per wave32
    float o = orow[i];
    int   h = head[i];
    osum += o;
    if (h) ohead += o;
    unsigned long long bal = __ballot(o > 0.0f);
    if ((tid & 31) == 0) omask[i >> 5] = (unsigned)bal;
  }
  osum  = block_reduce_add(osum,  redf, tid);
  ohead = block_reduce_add(ohead, redf, tid);

  // wait for async row, make LDS visible to all waves
  asm volatile("s_wait_asynccnt 0x0" ::: "memory");
  __syncthreads();

  // ---- Pass 1b: score stats from LDS ----
  float ssum = 0.f, smin = INFINITY, smax = -INFINITY;
  for (int it = 0; it < PER; ++it) {
    float s = sn[it * TPB + tid];
    ssum += s; smin = fminf(smin, s); smax = fmaxf(smax, s);
  }
  ssum = block_reduce_add(ssum, redf, tid);
  smin = block_reduce_min(smin, redf, tid);
  smax = block_reduce_max(smax, redf, tid);

  bool skip = (ssum == 0.0f);                       // uniform across block
  if (skip) {
    for (int it = 0; it < PER; ++it) { int i = it * TPB + tid; wrow[i] = sn[i]; }
    return;
  }

  // ---- Pass 2: normalize + mask + histogram ----
  float range = smax - smin;
  for (int it = 0; it < PER; ++it) {
    int   i = it * TPB + tid;
    float v = (sn[i] - smin) / range;               // IEEE div, matches reference
    bool masked = (omask[i >> 5] >> (i & 31)) & 1u;
    if (masked) v = -10000.0f;
    sn[i] = v;
    int b = (v <= 0.0f) ? 0 : min(NBINS - 1, (int)(v * (float)NBINS));
    atomicAdd(&hist[b], 1u);
  }
  __syncthreads();

  // ---- Pass 3: largest bin B with count(bin >= B) >= RMAXC ----
  {
    unsigned csum = 0;
    for (int b = tid * 16; b < tid * 16 + 16; ++b) csum += hist[b];
    redi[tid] = (int)csum;
    __syncthreads();
    if (tid == 0) {
      int acc = 0, B = 0;
      for (int c = TPB - 1; c >= 0; --c) {
        if (acc + redi[c] >= RMAXC) {
          for (int b = c * 16 + 15; b >= c * 16; --b) {
            acc += (int)hist[b];
            if (acc >= RMAXC) { B = b; break; }
          }
          break;
        }
        acc += redi[c];
      }
      S->thrbin = B;
    }
    __syncthreads();
  }

  // ---- Pass 4: collect candidates in bins >= threshold ----
  {
    int B = S->thrbin;
    for (int it = 0; it < PER; ++it) {
      int   i = it * TPB + tid;
      float v = sn[i];
      int b = (v <= 0.0f) ? 0 : min(NBINS - 1, (int)(v * (float)NBINS));
      if (b >= B) {
        unsigned p = atomicAdd(&S->ccount, 1u);
        if (p < CAP) { cval[p] = v; cidx[p] = i; }
      }
    }
    __syncthreads();
  }
  int nc = (int)S->ccount; if (nc > CAP) nc = CAP;
  __syncthreads();

  // ---- Pass 5: exact rank (desc value, asc index on ties) -> top RMAXC ----
  for (int c = tid; c < nc; c += TPB) {
    float v = cval[c]; int id = cidx[c];
    int r = 0;
    for (int j = 0; j < nc; ++j) {
      float vj = cval[j];
      r += (vj > v) || (vj == v && cidx[j] < id);
    }
    if (r < RMAXC) { sval[r] = v; sidx[r] = id; }
  }
  __syncthreads();

  // ---- Pass 6: greedy diversity-aware selection (k steps) ----
  float p_head = ohead / osum;
  float p_long = (osum - ohead) / osum;
  float cv = 0.f, ch = 0.f;
  bool  mymask = false;
  if (tid < RMAXC) {
    cv = sval[tid];
    ch = head[sidx[tid]] ? 1.0f : 0.0f;
  }
  float c_head = 0.f, c_long = 0.f;
  for (int t = 0; t < KSEL; ++t) {
    float tf = (float)t;
    float ies_h = (t >= 2) ? 1.0f - c_head / fmaxf(tf, 1.0f) : 1.0f;
    float ies_l = (t >= 2) ? 1.0f - c_long / fmaxf(tf, 1.0f) : 1.0f;
    float comb;
    if (tid < RMAXC && !mymask) {
      float w = p_head * ch * ies_h + p_long * (1.0f - ch) * ies_l;
      comb = (1.0f - LMB) * cv + LMB * w;
    } else {
      comb = -INFINITY;
    }
    redf[tid] = comb; redi[tid] = tid;
    __syncthreads();
    for (int s = TPB / 2; s > 0; s >>= 1) {
      if (tid < s) {
        float a = redf[tid], b = redf[tid + s];
        int   ib = redi[tid + s];
        if (b > a || (b == a && ib < redi[tid])) { redf[tid] = b; redi[tid] = ib; }
      }
      __syncthreads();
    }
    int pick = redi[0];
    if (tid == pick) { S->pickch = ch; S->sel[t] = sidx[tid]; mymask = true; }
    __syncthreads();
    c_head += S->pickch;
    c_long += 1.0f - S->pickch;
    __syncthreads();
  }

  // ---- Pass 7: write output row (hazard-free: compare against sel ids) ----
  int selreg[KSEL];
  #pragma unroll
  for (int j = 0; j < KSEL; ++j) selreg[j] = S->sel[j];
  for (int it = 0; it < PER; ++it) {
    int i = it * TPB + tid;
    float v = 0.0f;
    #pragma unroll
    for (int j = 0; j < KSEL; ++j)
      if (i == selreg[j]) v = (float)(KSEL - 1 - j) / (float)KSEL;
    wrow[i] = v;
  }
}

extern "C" void kernel_launch(void* const* d_in, const int* in_sizes, int n_in,
                              void* d_out, int out_size, void* d_ws, size_t ws_size,
                              hipStream_t stream) {
  const float* pred = (const float*)d_in[0];
  const float* obs  = (const float*)d_in[1];
  const int*   head = (const int*)d_in[2];
  float*       out  = (float*)d_out;
  int U = in_sizes[0] / NITEMS;   // 4096

  (void)d_ws; (void)ws_size; (void)n_in; (void)out_size;
  hipFuncSetAttribute((const void*)rerank_kernel,
                      hipFuncAttributeMaxDynamicSharedMemorySize, SMEM_BYTES);
  rerank_kernel<<<U, TPB, SMEM_BYTES, stream>>>(pred, obs, head, out);
}